// AngularPenaltySMLoss_10239202033884
// MI455X (gfx1250) — compile-verified
//
#include <hip/hip_runtime.h>
#include <hip/hip_bf16.h>

typedef __attribute__((ext_vector_type(2))) float v2f;
typedef __attribute__((ext_vector_type(8))) float v8f;

#define B_ROWS   1024
#define C_COLS   100000
#define S_SCALE  64.0f
#define COS_M    0.877582561890373f   // cos(0.5)
#define SIN_M    0.479425538604203f   // sin(0.5)
#define EPS_CL   1e-7f

// ---- CDNA5 async global->LDS helpers (gfx1250) -----------------------------
__device__ __forceinline__ unsigned lds_lo32(const void* p) {
  // Flat LDS addresses carry the wave-relative LDS byte offset in addr[31:0].
  return (unsigned)(uintptr_t)p;
}

__device__ __forceinline__ void async_copy_b128(unsigned lds_off, const void* gptr) {
  asm volatile("global_load_async_to_lds_b128 %0, %1, off"
               :: "v"(lds_off), "v"((unsigned long long)(uintptr_t)gptr)
               : "memory");
}

#define WAIT_ASYNCCNT_LE2() asm volatile("s_wait_asynccnt 0x2" ::: "memory")
#define WAIT_ASYNCCNT_0()   asm volatile("s_wait_asynccnt 0x0" ::: "memory")
#define WAIT_DSCNT_0()      asm volatile("s_wait_dscnt 0x0"    ::: "memory")

// ---- Kernel 1: gather target logit, compute numerator and exp(S*t) ---------
__global__ __launch_bounds__(256) void arc_gather_kernel(
    const float* __restrict__ pred, const int* __restrict__ labels,
    float* __restrict__ numv, float* __restrict__ expst) {
  int i = blockIdx.x * blockDim.x + threadIdx.x;
  if (i < B_ROWS) {
    int lbl = labels[i];
    float t = pred[(size_t)i * C_COLS + (size_t)lbl];
    expst[i] = __expf(S_SCALE * t);                      // uses UNclamped t (matches ref)
    float tc = fminf(fmaxf(t, -1.0f + EPS_CL), 1.0f - EPS_CL);
    // S * cos(acos(tc) + M) == S * (tc*cosM - sqrt(1-tc^2)*sinM)
    float s2 = fmaxf(1.0f - tc * tc, 0.0f);
    numv[i] = S_SCALE * (tc * COS_M - __fsqrt_rn(s2) * SIN_M);
  }
}

// ---- Kernel 2: per-row sum of exp(S*pred) via async-to-LDS pipeline --------
__global__ __launch_bounds__(256) void arc_rowsum_kernel(
    const float* __restrict__ pred, float* __restrict__ rowsum) {
  constexpr int DEPTH = 3;                       // 3-deep async pipeline
  constexpr int NIT   = C_COLS / 1024;           // 97 full tiles (256 thr * float4)
  constexpr int TAILV = (C_COLS - NIT * 1024) / 4; // 168 tail float4s

  __shared__ float4 buf[DEPTH][256];             // 12 KB
  __shared__ float  wsum[8];

  const int tid = threadIdx.x;
  const int row = blockIdx.x;
  const float* p = pred + (size_t)row * C_COLS;

  // Prologue: fill DEPTH-1 buffers.
  for (int k = 0; k < DEPTH - 1; ++k)
    async_copy_b128(lds_lo32(&buf[k][tid]), (const float4*)(p + k * 1024) + tid);

  float acc = 0.0f;
  for (int it = 0; it < NIT; ++it) {
    const int ahead = it + DEPTH - 1;
    if (ahead < NIT) {
      async_copy_b128(lds_lo32(&buf[ahead % DEPTH][tid]),
                      (const float4*)(p + ahead * 1024) + tid);
      WAIT_ASYNCCNT_LE2();                       // <=2 outstanding -> buffer `it` landed
    } else {
      WAIT_ASYNCCNT_0();
    }
    float4 v = buf[it % DEPTH][tid];             // ds_load_b128, own slot (no barrier)
    acc += __expf(S_SCALE * v.x) + __expf(S_SCALE * v.y);
    acc += __expf(S_SCALE * v.z) + __expf(S_SCALE * v.w);
    WAIT_DSCNT_0();                              // LDS read done before slot is re-issued
  }

  // Tail: 672 floats = 168 float4s, straight global loads.
  if (tid < TAILV) {
    float4 v = *((const float4*)(p + NIT * 1024) + tid);
    acc += __expf(S_SCALE * v.x) + __expf(S_SCALE * v.y);
    acc += __expf(S_SCALE * v.z) + __expf(S_SCALE * v.w);
  }

  // Wave32 reduce, then cross-wave via LDS.
  for (int off = 16; off > 0; off >>= 1) acc += __shfl_xor(acc, off, 32);
  if ((tid & 31) == 0) wsum[tid >> 5] = acc;
  __syncthreads();
  if (tid < 8) {
    float s = wsum[tid];
    for (int off = 4; off > 0; off >>= 1) s += __shfl_xor(s, off, 8);
    if (tid == 0) rowsum[row] = s;
  }
}

// ---- Kernel 3: per-row loss + WMMA reduction to -mean(L) -------------------
__global__ __launch_bounds__(32) void arc_finalize_kernel(
    const float* __restrict__ numv, const float* __restrict__ expst,
    const float* __restrict__ rowsum, float* __restrict__ out) {
  __shared__ float Lsh[B_ROWS];
  const int lane = threadIdx.x;

  for (int j = 0; j < B_ROWS / 32; ++j) {
    int i = j * 32 + lane;
    float num   = numv[i];
    float denom = __expf(num) + (rowsum[i] - expst[i]);
    Lsh[i] = num - __logf(denom);
  }
  __syncthreads();

  // Reduce 1024 values with v_wmma_f32_16x16x4_f32: D += A(16x4 chunk) * ones(4x16).
  // A layout: lane L -> row m = L&15; VGPR0/1 hold K = kb, kb+1 with kb = (L>>4)*2.
  v8f Cacc = {0.f, 0.f, 0.f, 0.f, 0.f, 0.f, 0.f, 0.f};
  const int m  = lane & 15;
  const int kb = (lane >> 4) << 1;
  const v2f Bones = {1.0f, 1.0f};
  for (int c = 0; c < 16; ++c) {
    v2f A;
    A.x = Lsh[c * 64 + m * 4 + kb];
    A.y = Lsh[c * 64 + m * 4 + kb + 1];
    Cacc = __builtin_amdgcn_wmma_f32_16x16x4_f32(
        /*neg_a=*/false, A, /*neg_b=*/false, Bones,
        /*c_mod=*/(short)0, Cacc, /*reuse_a=*/false, /*reuse_b=*/false);
  }
  // Each lane holds D[m..m+7][n] (one column); rows 0-7 in lanes 0-15, 8-15 in 16-31.
  float s = Cacc[0] + Cacc[1] + Cacc[2] + Cacc[3] +
            Cacc[4] + Cacc[5] + Cacc[6] + Cacc[7];
  s += __shfl_xor(s, 16, 32);                    // combine the two M-halves
  if (lane == 0) out[0] = -s * (1.0f / (float)B_ROWS);
}

// ---- Launch ----------------------------------------------------------------
extern "C" void kernel_launch(void* const* d_in, const int* in_sizes, int n_in,
                              void* d_out, int out_size, void* d_ws, size_t ws_size,
                              hipStream_t stream) {
  const float* pred   = (const float*)d_in[0];
  const int*   labels = (const int*)d_in[1];
  float*       out    = (float*)d_out;

  float* numv   = (float*)d_ws;          // [1024]
  float* expst  = numv  + B_ROWS;        // [1024]
  float* rowsum = expst + B_ROWS;        // [1024]

  arc_gather_kernel  <<<(B_ROWS + 255) / 256, 256, 0, stream>>>(pred, labels, numv, expst);
  arc_rowsum_kernel  <<<B_ROWS,               256, 0, stream>>>(pred, rowsum);
  arc_finalize_kernel<<<1,                     32, 0, stream>>>(numv, expst, rowsum, out);
  (void)in_sizes; (void)n_in; (void)out_size; (void)ws_size;
}